// MOEPred_42863773614422
// MI455X (gfx1250) — compile-verified
//
#include <hip/hip_runtime.h>
#include <hip/hip_bf16.h>

// MoE top-2 prediction head, bandwidth-bound streaming kernel for gfx1250.
// Combined GEMM: (B x 256) x (256 x 16) via V_WMMA_F32_16X16X4_F32, then
// per-row softmax/top-2/combine epilogue on 16 lanes per wave.

typedef __attribute__((ext_vector_type(2))) float v2f;
typedef __attribute__((ext_vector_type(8))) float v8f;

#define D_DIM 256
#define NEXP 8
#define KSTEPS 64           // 256 / 4
#define TILES_PER_WAVE 8
#define WAVES_PER_BLOCK 8
#define EPAD 20             // epilogue row stride (floats): conflict-free b128 reads

__global__ __launch_bounds__(256)
void MOEPred_kernel(const float* __restrict__ X,
                    const float* __restrict__ Wg,
                    const float* __restrict__ bg,
                    const float* __restrict__ We,
                    const float* __restrict__ be,
                    float* __restrict__ out)
{
    // B-fragments pre-packed in WMMA lane order: s_wb[step*64 + lane*2 + {0,1}]
    __shared__ float s_wb[KSTEPS * 64];                    // 16 KB
    __shared__ float s_ep[WAVES_PER_BLOCK * 16 * EPAD];    // 10 KB epilogue scratch

    const int tid  = threadIdx.x;
    const int wave = tid >> 5;
    const int lane = tid & 31;

    // ---- Stage combined weight matrix W[k][n] (n<8: Wg, n>=8: We^T) into LDS,
    // already permuted into per-lane WMMA B fragments.
    // B layout (32-bit, 4x16): lane l holds N = l&15, K = 2*(l>>4)+{0,1}.
    #pragma unroll
    for (int j = 0; j < 8; ++j) {
        int idx = j * 256 + tid;               // 0..2047 fragment slots
        int s   = idx >> 5;
        int l   = idx & 31;
        int k0  = 4 * s + 2 * (l >> 4);
        int n   = l & 15;
        float wx, wy;
        if (n < NEXP) {
            wx = Wg[(size_t)k0 * NEXP + n];
            wy = Wg[(size_t)(k0 + 1) * NEXP + n];
        } else {
            const float* wr = We + (size_t)(n - NEXP) * D_DIM;
            wx = wr[k0];
            wy = wr[k0 + 1];
        }
        s_wb[idx * 2]     = wx;
        s_wb[idx * 2 + 1] = wy;
    }

    // Biases (tiny, L2-resident) into registers.
    float bgr[NEXP], ber[NEXP];
    #pragma unroll
    for (int n = 0; n < NEXP; ++n) { bgr[n] = bg[n]; ber[n] = be[n]; }

    __syncthreads();

    const int half = lane >> 4;     // which K-half of the A fragment this lane holds
    const int col  = lane & 15;     // A row within tile / D column
    const int epb  = wave * (16 * EPAD);

    for (int t = 0; t < TILES_PER_WAVE; ++t) {
        const int tile = (blockIdx.x * WAVES_PER_BLOCK + wave) * TILES_PER_WAVE + t;

        // A fragment source: lane l streams row (tile*16 + l%16), 8B per step.
        const float* gA = X + (size_t)tile * (16 * D_DIM)
                            + (size_t)col * D_DIM + 2 * half;

        v8f acc = {};
        #pragma unroll 8
        for (int s = 0; s < KSTEPS; ++s) {
            v2f a = *(const v2f*)(gA + 4 * s);                 // global_load_b64
            v2f b = *(const v2f*)(&s_wb[s * 64 + 2 * lane]);   // ds_load_b64
            acc = __builtin_amdgcn_wmma_f32_16x16x4_f32(
                      /*neg_a=*/false, a, /*neg_b=*/false, b,
                      /*c_mod=*/(short)0, acc,
                      /*reuse_a=*/false, /*reuse_b=*/false);
        }

        // ---- Transpose D (cols live across lanes) through LDS:
        // D layout: VGPR j = row j (lanes 0-15) / row j+8 (lanes 16-31), N = lane&15.
        #pragma unroll
        for (int j = 0; j < 8; ++j)
            s_ep[epb + (j + 8 * half) * EPAD + col] = acc[j];

        asm volatile("s_wait_dscnt 0" ::: "memory");

        if (lane < 16) {
            const float* r = &s_ep[epb + lane * EPAD];
            float v[16];
            #pragma unroll
            for (int n = 0; n < 16; ++n) v[n] = r[n];

            // Top-2 over gate logits; softmax(topk(softmax(z))) == sigmoid gap.
            float z0 = -3.4e38f, z1 = -3.4e38f, ya = 0.f, yb = 0.f;
            #pragma unroll
            for (int n = 0; n < NEXP; ++n) {
                float g = v[n] + bgr[n];            // gate logit
                float y = v[NEXP + n] + ber[n];     // expert output
                bool gt0 = g > z0;
                bool gt1 = g > z1;
                float nz1 = gt0 ? z0 : (gt1 ? g : z1);
                float ny1 = gt0 ? ya : (gt1 ? y : yb);
                z1 = nz1; yb = ny1;
                z0 = gt0 ? g : z0;
                ya = gt0 ? y : ya;
            }
            float e  = __expf(z1 - z0);
            float w1 = e / (1.0f + e);
            float w0 = 1.0f - w1;
            out[(size_t)tile * 16 + lane] = w0 * ya + w1 * yb;
        }
    }
}

extern "C" void kernel_launch(void* const* d_in, const int* in_sizes, int n_in,
                              void* d_out, int out_size, void* d_ws, size_t ws_size,
                              hipStream_t stream) {
    const float* X  = (const float*)d_in[0];
    const float* Wg = (const float*)d_in[1];
    const float* bg = (const float*)d_in[2];
    const float* We = (const float*)d_in[3];
    const float* be = (const float*)d_in[4];
    float* out = (float*)d_out;

    const int rows   = out_size;                       // B = 1048576
    const int tiles  = rows / 16;                      // 65536
    const int blocks = tiles / (WAVES_PER_BLOCK * TILES_PER_WAVE);  // 1024

    MOEPred_kernel<<<blocks, 256, 0, stream>>>(X, Wg, bg, We, be, out);
}